// Contrast_43404939493422
// MI455X (gfx1250) — compile-verified
//
#include <hip/hip_runtime.h>
#include <hip/hip_bf16.h>

typedef float v2f __attribute__((ext_vector_type(2)));
typedef float v8f __attribute__((ext_vector_type(8)));

#define N_ROWS 8192
#define DIM 64
#define JCHUNKS 8               // j-dimension split into 8 chunks of 64 tiles
#define NTILES (N_ROWS / 16)    // 512
#define TILES_PER_CHUNK (NTILES / JCHUNKS)  // 64

__device__ __forceinline__ v8f wmma_f32(v2f a, v2f b, v8f c) {
    // D = A(16x4 f32) * B(4x16 f32) + C(16x16 f32)
    return __builtin_amdgcn_wmma_f32_16x16x4_f32(
        false, a, false, b, (short)0, c, false, false);
}

// ---------------------------------------------------------------------------
// Kernel 1: row-wise L2 normalize (one wave32 per row of 64 floats)
// ---------------------------------------------------------------------------
__global__ void __launch_bounds__(256)
norm_kernel(const float* __restrict__ z, float* __restrict__ zn) {
    const int gid  = blockIdx.x * blockDim.x + threadIdx.x;
    const int row  = gid >> 5;
    const int lane = threadIdx.x & 31;
    const float* rp = z + (size_t)row * DIM;
    v2f v = *(const v2f*)(rp + lane * 2);
    float s = v.x * v.x + v.y * v.y;
    #pragma unroll
    for (int m = 16; m >= 1; m >>= 1) s += __shfl_xor(s, m, 32);
    float nrm = sqrtf(s);
    float inv = 1.0f / fmaxf(nrm, 1e-12f);
    v2f o; o.x = v.x * inv; o.y = v.y * inv;
    *(v2f*)(zn + (size_t)row * DIM + lane * 2) = o;
}

// ---------------------------------------------------------------------------
// Kernel 2: fused 4-way similarity + exp + masked row sums.
// One wave handles one 16-row i-tile over one j-chunk (64 j-tiles).
// Needs ~200 VGPRs resident -> explicitly allow low occupancy (no spills).
// pos (256 MB, streamed once) is loaded non-temporally to keep L2 for z data.
// ---------------------------------------------------------------------------
__global__ void __launch_bounds__(256)
__attribute__((amdgpu_waves_per_eu(1, 4)))
contrast_kernel(const float* __restrict__ zmp,
                const float* __restrict__ zsc,
                const float* __restrict__ pos,
                float* __restrict__ rowsum_buf,
                float* __restrict__ possum_buf) {
    const int gtid    = blockIdx.x * blockDim.x + threadIdx.x;
    const int pairIdx = gtid >> 5;               // global wave id
    const int itile   = pairIdx >> 3;            // / JCHUNKS
    const int jchunk  = pairIdx & (JCHUNKS - 1);
    const int lane    = threadIdx.x & 31;
    const int m       = lane & 15;               // row-within-tile (A) / col (B)
    const int h2      = (lane >> 4) << 1;        // 0 or 2: K-pair select

    // Resident A fragments: rows itile*16+m of normalized mp and sc, full D.
    const float* amp_row = zmp + (size_t)(itile * 16 + m) * DIM;
    const float* asc_row = zsc + (size_t)(itile * 16 + m) * DIM;
    v2f amp[16], asc[16];
    #pragma unroll
    for (int kk = 0; kk < 16; ++kk) {
        amp[kk] = *(const v2f*)(amp_row + kk * 4 + h2);
        asc[kk] = *(const v2f*)(asc_row + kk * 4 + h2);
    }

    // Per-row partial accumulators (row M = r + 8*(lane>=16) of the i-tile),
    // this lane covers column j = jt*16 + m within each tile.
    float rs0[8], rs1[8], rs2[8], rs3[8];
    float ps0[8], ps1[8], ps2[8], ps3[8];
    #pragma unroll
    for (int r = 0; r < 8; ++r) {
        rs0[r] = rs1[r] = rs2[r] = rs3[r] = 0.0f;
        ps0[r] = ps1[r] = ps2[r] = ps3[r] = 0.0f;
    }

    const int jt0 = jchunk * TILES_PER_CHUNK;
    const int jt1 = jt0 + TILES_PER_CHUNK;
    const size_t pos_row0 = (size_t)(itile * 16 + (lane >> 4) * 8) * N_ROWS;

    for (int jt = jt0; jt < jt1; ++jt) {
        const float* bmp_row = zmp + (size_t)(jt * 16 + m) * DIM;
        const float* bsc_row = zsc + (size_t)(jt * 16 + m) * DIM;

        v8f c0 = {}; v8f c1 = {}; v8f c2 = {}; v8f c3 = {};
        #pragma unroll
        for (int kk = 0; kk < 16; ++kk) {
            v2f bmp = *(const v2f*)(bmp_row + kk * 4 + h2);
            v2f bsc = *(const v2f*)(bsc_row + kk * 4 + h2);
            c0 = wmma_f32(amp[kk], bsc, c0);  // mp2sc
            c1 = wmma_f32(asc[kk], bmp, c1);  // sc2mp
            c2 = wmma_f32(amp[kk], bmp, c2);  // mp2mp
            c3 = wmma_f32(asc[kk], bsc, c3);  // sc2sc
        }

        const float* prow = pos + pos_row0 + (size_t)(jt * 16 + m);
        #pragma unroll
        for (int r = 0; r < 8; ++r) {
            // Streamed exactly once -> non-temporal (TH=NT) load.
            float p = __builtin_nontemporal_load(prow + (size_t)r * N_ROWS);
            float e;
            e = __expf(c0[r] * 2.0f); rs0[r] += e; ps0[r] += e * p;
            e = __expf(c1[r] * 2.0f); rs1[r] += e; ps1[r] += e * p;
            e = __expf(c2[r] * 2.0f); rs2[r] += e; ps2[r] += e * p;
            e = __expf(c3[r] * 2.0f); rs3[r] += e; ps3[r] += e * p;
        }
    }

    // Reduce over the 16 lanes of each half (columns 0..15 of the tiles).
    #pragma unroll
    for (int r = 0; r < 8; ++r) {
        #pragma unroll
        for (int msk = 8; msk >= 1; msk >>= 1) {
            rs0[r] += __shfl_xor(rs0[r], msk, 32);
            rs1[r] += __shfl_xor(rs1[r], msk, 32);
            rs2[r] += __shfl_xor(rs2[r], msk, 32);
            rs3[r] += __shfl_xor(rs3[r], msk, 32);
            ps0[r] += __shfl_xor(ps0[r], msk, 32);
            ps1[r] += __shfl_xor(ps1[r], msk, 32);
            ps2[r] += __shfl_xor(ps2[r], msk, 32);
            ps3[r] += __shfl_xor(ps3[r], msk, 32);
        }
    }

    if ((lane & 15) == 0) {
        const int half = lane >> 4;              // rows r (half 0) / 8+r (half 1)
        const size_t tilebase = (size_t)(itile * JCHUNKS + jchunk) * 4 * 16;
        #pragma unroll
        for (int r = 0; r < 8; ++r) {
            size_t i0 = tilebase + 0 * 16 + half * 8 + r;
            size_t i1 = tilebase + 1 * 16 + half * 8 + r;
            size_t i2 = tilebase + 2 * 16 + half * 8 + r;
            size_t i3 = tilebase + 3 * 16 + half * 8 + r;
            rowsum_buf[i0] = rs0[r]; possum_buf[i0] = ps0[r];
            rowsum_buf[i1] = rs1[r]; possum_buf[i1] = ps1[r];
            rowsum_buf[i2] = rs2[r]; possum_buf[i2] = ps2[r];
            rowsum_buf[i3] = rs3[r]; possum_buf[i3] = ps3[r];
        }
    }
}

// ---------------------------------------------------------------------------
// Kernel 3: deterministic final reduction (single block).
// ---------------------------------------------------------------------------
__global__ void __launch_bounds__(256)
finalize_kernel(const float* __restrict__ rowsum_buf,
                const float* __restrict__ possum_buf,
                float* __restrict__ out) {
    __shared__ float red[256];
    float local = 0.0f;
    for (int idx = threadIdx.x; idx < N_ROWS * 4; idx += 256) {
        const int mat   = idx & 3;
        const int row   = idx >> 2;
        const int itile = row >> 4;
        const int rowin = row & 15;
        float rsum = 0.0f, psum = 0.0f;
        for (int ch = 0; ch < JCHUNKS; ++ch) {
            size_t base = ((size_t)(itile * JCHUNKS + ch) * 4 + mat) * 16 + rowin;
            rsum += rowsum_buf[base];
            psum += possum_buf[base];
        }
        local += -logf(psum / (rsum + 1e-8f));
    }
    red[threadIdx.x] = local;
    __syncthreads();
    for (int s = 128; s > 0; s >>= 1) {
        if ((int)threadIdx.x < s) red[threadIdx.x] += red[threadIdx.x + s];
        __syncthreads();
    }
    if (threadIdx.x == 0) out[0] = red[0] / (float)N_ROWS;
}

extern "C" void kernel_launch(void* const* d_in, const int* in_sizes, int n_in,
                              void* d_out, int out_size, void* d_ws, size_t ws_size,
                              hipStream_t stream) {
    const float* z_mp = (const float*)d_in[0];
    const float* z_sc = (const float*)d_in[1];
    const float* pos  = (const float*)d_in[2];
    float* out = (float*)d_out;

    float* ws = (float*)d_ws;
    float* zmp_n      = ws;                                  // 8192*64
    float* zsc_n      = zmp_n + (size_t)N_ROWS * DIM;        // 8192*64
    float* rowsum_buf = zsc_n + (size_t)N_ROWS * DIM;        // 512*8*4*16 = 262144
    float* possum_buf = rowsum_buf + (size_t)NTILES * JCHUNKS * 4 * 16;

    // 1 wave per row: 8192 rows * 32 threads / 256 = 1024 blocks
    norm_kernel<<<1024, 256, 0, stream>>>(z_mp, zmp_n);
    norm_kernel<<<1024, 256, 0, stream>>>(z_sc, zsc_n);

    // 512 itiles * 8 jchunks = 4096 waves; 8 waves/block -> 512 blocks
    contrast_kernel<<<512, 256, 0, stream>>>(zmp_n, zsc_n, pos,
                                             rowsum_buf, possum_buf);

    finalize_kernel<<<1, 256, 0, stream>>>(rowsum_buf, possum_buf, out);
}